// YoloLoss_26938034880867
// MI455X (gfx1250) — compile-verified
//
#include <hip/hip_runtime.h>

typedef float v4f __attribute__((ext_vector_type(4)));
typedef float v2f __attribute__((ext_vector_type(2)));
typedef float v8f __attribute__((ext_vector_type(8)));

#define LAMBDA_COORD 5.0f
#define LAMBDA_NOOBJ 0.5f

// Compile-time block geometry so the LDS combine fully unrolls.
#define THREADS 256
#define NWAVES  (THREADS / 32)

__device__ __forceinline__ float sq(float x) { return x * x; }

// Single-instruction xor-lane-swap via ds_swizzle (group-of-32 mode):
// pattern = (xor_mask << 10) | (or_mask << 5) | and_mask(0x1f)
template <int PAT>
__device__ __forceinline__ float swz_add(float s) {
    int v = __builtin_amdgcn_ds_swizzle(__float_as_int(s), PAT);
    return s + __int_as_float(v);
}

// Block-level sum reduction. Wave level goes through V_WMMA_F32_16X16X4_F32:
//   A = all-ones (layout-independent: all 64 A slots = 1.0),
//   B = {partial, 0} per lane (the 2 B VGPRs map bijectively onto the 64
//       elements of the 4x16 B matrix -> sum of all B elements == wave sum,
//       independent of the B striping),
//   C = 0  =>  D[m][n] = colsum(B)[n] for every m.
// Documented C/D layout: d[0] = D[0][n] (lanes 0-15) / D[8][n] (lanes 16-31),
// both equal to colsums -> xor-reduce d[0] over 32 lanes counts each colsum
// twice; halve to get the wave sum. All threads must execute this (EXEC=all-1).
__device__ __forceinline__ float block_reduce_sum(float partial, float* lds) {
    v2f a; a.x = 1.0f;    a.y = 1.0f;
    v2f b; b.x = partial; b.y = 0.0f;
    v8f c = {};
    v8f d = __builtin_amdgcn_wmma_f32_16x16x4_f32(
        false, a, false, b, (short)0, c, false, false);

    float s = d[0];
    s = swz_add<0x401F>(s);   // xor 16
    s = swz_add<0x201F>(s);   // xor 8
    s = swz_add<0x101F>(s);   // xor 4
    s = swz_add<0x081F>(s);   // xor 2
    s = swz_add<0x041F>(s);   // xor 1
    s *= 0.5f;                // each colsum appears twice in d[0]

    const int wid  = threadIdx.x >> 5;
    const int lane = threadIdx.x & 31;
    if (lane == 0) lds[wid] = s;
    __syncthreads();

    float tot = 0.0f;
    if (threadIdx.x == 0) {
        #pragma unroll
        for (int i = 0; i < NWAVES; ++i) tot += lds[i];  // fixed order: deterministic
    }
    return tot;   // valid on thread 0 only
}

// Per-cell YOLO loss given the first 10 pred/target floats and the raw
// (unweighted) sum of squared class diffs (indices 10..29).
__device__ __forceinline__ float cell_loss(const float* p, const float* t,
                                           float l5raw) {
    const float obj   = (t[4] >  0.0f) ? 1.0f : 0.0f;
    const float noobj = (t[4] == 0.0f) ? 1.0f : 0.0f;

    const float a2 = (t[2] - t[0]) * (t[3] - t[1]);

    // IoU pred box 1 (p0..p3) vs target box (t0..t3)
    float ltx = fmaxf(p[0], t[0]), lty = fmaxf(p[1], t[1]);
    float rbx = fminf(p[2], t[2]), rby = fminf(p[3], t[3]);
    float w   = fmaxf(rbx - ltx, 0.0f), h = fmaxf(rby - lty, 0.0f);
    float inter1 = w * h;
    float a1     = (p[2] - p[0]) * (p[3] - p[1]);
    float iou1   = inter1 / (a1 + a2 - inter1);

    // IoU pred box 2 (p5..p8) vs target box
    ltx = fmaxf(p[5], t[0]); lty = fmaxf(p[6], t[1]);
    rbx = fminf(p[7], t[2]); rby = fminf(p[8], t[3]);
    w   = fmaxf(rbx - ltx, 0.0f); h = fmaxf(rby - lty, 0.0f);
    float inter2 = w * h;
    float a1b    = (p[7] - p[5]) * (p[8] - p[6]);
    float iou2   = inter2 / (a1b + a2 - inter2);

    const bool pick2 = (iou2 > iou1);   // argmax, first max wins

    float l1, l2, conf;
    if (pick2) {
        l1   = sq(p[5] - t[5]) + sq(p[6] - t[6]);
        l2   = sq(sqrtf(p[7]) - sqrtf(t[7])) + sq(sqrtf(p[8]) - sqrtf(t[8]));
        conf = sq(p[9] - t[9]);
    } else {
        l1   = sq(p[0] - t[0]) + sq(p[1] - t[1]);
        l2   = sq(sqrtf(p[2]) - sqrtf(t[2])) + sq(sqrtf(p[3]) - sqrtf(t[3]));
        conf = sq(p[4] - t[4]);
    }

    return LAMBDA_COORD * obj * (l1 + l2)
         + conf * (obj + LAMBDA_NOOBJ * noobj)
         + obj * l5raw;
}

// One thread processes a PAIR of cells: 2*30 floats = 240 B per tensor,
// 16-byte aligned -> 15 aligned b128 non-temporal loads per tensor.
// Class scores (idx 10..29 of each cell) are folded into a running
// sum-of-squares during the load loop (compile-time dispatch), so only
// p[0..9]/t[0..9] of each cell stay live.
__global__ void __launch_bounds__(THREADS)
yolo_loss_partial(const float* __restrict__ pred,
                  const float* __restrict__ targ,
                  float* __restrict__ block_sums,
                  int n_pairs) {
    __shared__ float lds[NWAVES];

    const int pair = blockIdx.x * THREADS + threadIdx.x;
    float partial = 0.0f;

    if (pair < n_pairs) {
        const v4f* p4 = (const v4f*)pred + (size_t)pair * 15;
        const v4f* t4 = (const v4f*)targ + (size_t)pair * 15;

        float pa[10], ta[10], pb[10], tb[10];
        float l5a = 0.0f, l5b = 0.0f;

        #pragma unroll
        for (int j = 0; j < 15; ++j) {
            const v4f pv = __builtin_nontemporal_load(p4 + j);
            const v4f tv = __builtin_nontemporal_load(t4 + j);
            #pragma unroll
            for (int k = 0; k < 4; ++k) {
                const int idx = 4 * j + k;          // 0..59, compile-time
                const float pe = pv[k], te = tv[k];
                if (idx < 10) {
                    pa[idx] = pe; ta[idx] = te;
                } else if (idx < 30) {
                    const float d = pe - te; l5a += d * d;
                } else if (idx < 40) {
                    pb[idx - 30] = pe; tb[idx - 30] = te;
                } else {
                    const float d = pe - te; l5b += d * d;
                }
            }
        }

        partial = cell_loss(pa, ta, l5a) + cell_loss(pb, tb, l5b);
    }

    const float bs = block_reduce_sum(partial, lds);
    if (threadIdx.x == 0) block_sums[blockIdx.x] = bs;
}

__global__ void __launch_bounds__(THREADS)
yolo_loss_final(const float* __restrict__ block_sums,
                int n, float* __restrict__ out) {
    __shared__ float lds[NWAVES];
    float acc = 0.0f;
    for (int i = threadIdx.x; i < n; i += THREADS)
        acc += block_sums[i];                     // fixed strided order: deterministic
    const float tot = block_reduce_sum(acc, lds);
    if (threadIdx.x == 0) out[0] = tot;
}

extern "C" void kernel_launch(void* const* d_in, const int* in_sizes, int n_in,
                              void* d_out, int out_size, void* d_ws, size_t ws_size,
                              hipStream_t stream) {
    (void)n_in; (void)out_size; (void)ws_size;
    const float* pred = (const float*)d_in[0];
    const float* targ = (const float*)d_in[1];
    float* out        = (float*)d_out;
    float* bsums      = (float*)d_ws;

    const int total_elems = in_sizes[0];          // 2048*28*28*30
    const int n_cells     = total_elems / 30;     // 1,605,632
    const int n_pairs     = n_cells / 2;          // 802,816
    const int blocks      = (n_pairs + THREADS - 1) / THREADS;  // 3136

    yolo_loss_partial<<<blocks, THREADS, 0, stream>>>(pred, targ, bsums, n_pairs);
    yolo_loss_final<<<1, THREADS, 0, stream>>>(bsums, blocks, out);
}